// CoupledFEMSolver_12017318494905
// MI455X (gfx1250) — compile-verified
//
#include <hip/hip_runtime.h>
#include <hip/hip_bf16.h>

#define N_Fc 9000
#define N_Sc 3000
#define EFc 250000
#define ESc 80000
#define NDOFc 9000  // 3 * N_S

typedef __attribute__((ext_vector_type(2))) float v2f;
typedef __attribute__((ext_vector_type(4))) float f4;
typedef __attribute__((ext_vector_type(8))) float v8f;

// ---------------------------------------------------------------------------
// Zero-fill 648 MB of output with non-temporal 128-bit stores (pure stream).
// ---------------------------------------------------------------------------
__global__ void __launch_bounds__(256)
zero_out_kernel(f4* __restrict__ p, int n4) {
  int i = blockIdx.x * blockDim.x + threadIdx.x;
  const int stride = gridDim.x * blockDim.x;
  f4 z = {0.f, 0.f, 0.f, 0.f};
  for (; i < n4; i += stride) {
    __builtin_nontemporal_store(z, p + i);
  }
}

// ---------------------------------------------------------------------------
// Fluid assembly: one tet element per lane. Closed-form tet gradients
// (cross products / signed det), 16 scattered f32 atomics per element.
// ---------------------------------------------------------------------------
__global__ void __launch_bounds__(256)
fluid_assemble_kernel(const float* __restrict__ nodes_f,
                      const int* __restrict__ elems,
                      float* __restrict__ Af, float kf2) {
  const int e = blockIdx.x * blockDim.x + threadIdx.x;
  if (e >= EFc) return;

  const int nd0 = elems[4 * e + 0];
  const int nd1 = elems[4 * e + 1];
  const int nd2 = elems[4 * e + 2];
  const int nd3 = elems[4 * e + 3];

  const float p0x = nodes_f[3 * nd0 + 0], p0y = nodes_f[3 * nd0 + 1], p0z = nodes_f[3 * nd0 + 2];
  const float p1x = nodes_f[3 * nd1 + 0], p1y = nodes_f[3 * nd1 + 1], p1z = nodes_f[3 * nd1 + 2];
  const float p2x = nodes_f[3 * nd2 + 0], p2y = nodes_f[3 * nd2 + 1], p2z = nodes_f[3 * nd2 + 2];
  const float p3x = nodes_f[3 * nd3 + 0], p3y = nodes_f[3 * nd3 + 1], p3z = nodes_f[3 * nd3 + 2];

  const float e1x = p1x - p0x, e1y = p1y - p0y, e1z = p1z - p0z;
  const float e2x = p2x - p0x, e2y = p2y - p0y, e2z = p2z - p0z;
  const float e3x = p3x - p0x, e3y = p3y - p0y, e3z = p3z - p0z;

  // cross(e2, e3)
  const float c1x = e2y * e3z - e2z * e3y;
  const float c1y = e2z * e3x - e2x * e3z;
  const float c1z = e2x * e3y - e2y * e3x;
  const float det = e1x * c1x + e1y * c1y + e1z * c1z;
  const float vol = fabsf(det) * (1.f / 6.f);
  const float inv = 1.f / det;

  float gx[4], gy[4], gz[4];
  gx[1] = c1x * inv; gy[1] = c1y * inv; gz[1] = c1z * inv;
  gx[2] = (e3y * e1z - e3z * e1y) * inv;
  gy[2] = (e3z * e1x - e3x * e1z) * inv;
  gz[2] = (e3x * e1y - e3y * e1x) * inv;
  gx[3] = (e1y * e2z - e1z * e2y) * inv;
  gy[3] = (e1z * e2x - e1x * e2z) * inv;
  gz[3] = (e1x * e2y - e1y * e2x) * inv;
  gx[0] = -(gx[1] + gx[2] + gx[3]);
  gy[0] = -(gy[1] + gy[2] + gy[3]);
  gz[0] = -(gz[1] + gz[2] + gz[3]);

  const int nd[4] = {nd0, nd1, nd2, nd3};
  const float c10 = vol * 0.1f;  // vol/10

#pragma unroll
  for (int i = 0; i < 4; ++i) {
#pragma unroll
    for (int j = 0; j < 4; ++j) {
      const float ke = vol * (gx[i] * gx[j] + gy[i] * gy[j] + gz[i] * gz[j]);
      const float me = c10 * ((i == j) ? 3.f : 1.f);  // 2I + ones
      atomicAdd(&Af[(size_t)nd[i] * N_Fc + nd[j]], ke - kf2 * me);
    }
  }
}

// ---------------------------------------------------------------------------
// Solid assembly: one element per wave32; Ke = vol * B^T (12x6) * (D*B)(6x12)
// computed with v_wmma_f32_16x16x4_f32 (K=6 padded to two k=4 steps).
// ---------------------------------------------------------------------------

// 12-way register select (branchless cndmask chain)
__device__ __forceinline__ float sel12(const float g[12], int idx) {
  float r = g[0];
#pragma unroll
  for (int t = 1; t < 12; ++t) r = (idx == t) ? g[t] : r;
  return r;
}

// Strain-displacement matrix entry B[k][c], c = 3*i + d, from gradients g[3i+d]
__device__ __forceinline__ float bval(const float g[12], int k, int c) {
  const int i = c / 3;
  const int d = c - 3 * i;
  const float gxv = sel12(g, 3 * i + 0);
  const float gyv = sel12(g, 3 * i + 1);
  const float gzv = sel12(g, 3 * i + 2);
  const float r0 = (d == 0) ? gxv : 0.f;
  const float r1 = (d == 1) ? gyv : 0.f;
  const float r2 = (d == 2) ? gzv : 0.f;
  const float r3 = (d == 0) ? gyv : ((d == 1) ? gxv : 0.f);
  const float r4 = (d == 1) ? gzv : ((d == 2) ? gyv : 0.f);
  const float r5 = (d == 0) ? gzv : ((d == 2) ? gxv : 0.f);
  return (k == 0) ? r0 : (k == 1) ? r1 : (k == 2) ? r2
       : (k == 3) ? r3 : (k == 4) ? r4 : (k == 5) ? r5 : 0.f;
}

// (D*B)[k][c] using the isotropic-D structure
__device__ __forceinline__ float dbval(const float g[12], int k, int c,
                                       float aD, float bD, float gD) {
  const float B0 = bval(g, 0, c);
  const float B1 = bval(g, 1, c);
  const float B2 = bval(g, 2, c);
  const float Bk = bval(g, k, c);            // 0 for k >= 6 (padding)
  const float s = B0 + B1 + B2;
  const float rlow = bD * s + (aD - bD) * Bk;
  return (k < 3) ? rlow : gD * Bk;           // k>=6 -> Bk==0 -> 0
}

__global__ void __launch_bounds__(256)
solid_assemble_kernel(const float* __restrict__ nodes_s,
                      const int* __restrict__ elems,
                      const float* __restrict__ Ep,
                      const float* __restrict__ nup,
                      const float* __restrict__ rhop,
                      float* __restrict__ As, float omega2) {
  const int lane = threadIdx.x & 31;
  const int e = (blockIdx.x * blockDim.x + threadIdx.x) >> 5;
  if (e >= ESc) return;  // wave-uniform: whole wave exits together

  const int n0 = elems[4 * e + 0];
  const int n1 = elems[4 * e + 1];
  const int n2 = elems[4 * e + 2];
  const int n3 = elems[4 * e + 3];

  const float p0x = nodes_s[3 * n0 + 0], p0y = nodes_s[3 * n0 + 1], p0z = nodes_s[3 * n0 + 2];
  const float p1x = nodes_s[3 * n1 + 0], p1y = nodes_s[3 * n1 + 1], p1z = nodes_s[3 * n1 + 2];
  const float p2x = nodes_s[3 * n2 + 0], p2y = nodes_s[3 * n2 + 1], p2z = nodes_s[3 * n2 + 2];
  const float p3x = nodes_s[3 * n3 + 0], p3y = nodes_s[3 * n3 + 1], p3z = nodes_s[3 * n3 + 2];

  const float e1x = p1x - p0x, e1y = p1y - p0y, e1z = p1z - p0z;
  const float e2x = p2x - p0x, e2y = p2y - p0y, e2z = p2z - p0z;
  const float e3x = p3x - p0x, e3y = p3y - p0y, e3z = p3z - p0z;

  const float c1x = e2y * e3z - e2z * e3y;
  const float c1y = e2z * e3x - e2x * e3z;
  const float c1z = e2x * e3y - e2y * e3x;
  const float det = e1x * c1x + e1y * c1y + e1z * c1z;
  const float vol = fabsf(det) * (1.f / 6.f);
  const float inv = 1.f / det;

  float g[12];  // g[3*i + d] = grad of node i, component d
  g[3]  = c1x * inv;
  g[4]  = c1y * inv;
  g[5]  = c1z * inv;
  g[6]  = (e3y * e1z - e3z * e1y) * inv;
  g[7]  = (e3z * e1x - e3x * e1z) * inv;
  g[8]  = (e3x * e1y - e3y * e1x) * inv;
  g[9]  = (e1y * e2z - e1z * e2y) * inv;
  g[10] = (e1z * e2x - e1x * e2z) * inv;
  g[11] = (e1x * e2y - e1y * e2x) * inv;
  g[0]  = -(g[3] + g[6] + g[9]);
  g[1]  = -(g[4] + g[7] + g[10]);
  g[2]  = -(g[5] + g[8] + g[11]);

  const float e0v = Ep[0];
  const float nuv = nup[0];
  const float rhov = rhop[0];
  const float coeff = e0v / ((1.f + nuv) * (1.f - 2.f * nuv));
  const float aD = coeff * (1.f - nuv);
  const float bD = coeff * nuv;
  const float gD = coeff * (1.f - 2.f * nuv) * 0.5f;

  const bool hi = (lane >= 16);
  const int mn = lane & 15;          // M for A-operand, N for B-operand
  const bool valid = (mn < 12);
  const int mnc = valid ? mn : 0;    // branchless clamp (keep EXEC intact pre-WMMA)

  // K assignment per lane-half and VGPR slot (A 16x4 / B 4x16 fp32 layouts)
  const int k0 = hi ? 2 : 0;
  const int k1 = hi ? 3 : 1;
  const int k2 = hi ? 6 : 4;  // k>=6 rows are zero padding
  const int k3 = hi ? 7 : 5;

  // A operand = B^T : A[m][k] = B[k][m]
  const float a0 = valid ? bval(g, k0, mnc) : 0.f;
  const float a1 = valid ? bval(g, k1, mnc) : 0.f;
  const float a2 = valid ? bval(g, k2, mnc) : 0.f;
  const float a3 = valid ? bval(g, k3, mnc) : 0.f;
  // B operand = D*B
  const float b0 = valid ? dbval(g, k0, mnc, aD, bD, gD) : 0.f;
  const float b1 = valid ? dbval(g, k1, mnc, aD, bD, gD) : 0.f;
  const float b2 = valid ? dbval(g, k2, mnc, aD, bD, gD) : 0.f;
  const float b3 = valid ? dbval(g, k3, mnc, aD, bD, gD) : 0.f;

  v2f A0 = {a0, a1};
  v2f A1 = {a2, a3};
  v2f B0 = {b0, b1};
  v2f B1 = {b2, b3};
  v8f C = {0.f, 0.f, 0.f, 0.f, 0.f, 0.f, 0.f, 0.f};
  // Ke_raw = B^T * (D*B), K=6 as two k=4 steps (upper k rows zero-padded)
  C = __builtin_amdgcn_wmma_f32_16x16x4_f32(false, A0, false, B0, (short)0, C,
                                            false, false);
  C = __builtin_amdgcn_wmma_f32_16x16x4_f32(false, A1, false, B1, (short)0, C,
                                            false, false);

  const float mterm = omega2 * rhov * vol * 0.25f;  // omega^2 * m_lump

  // Column DOF for this lane (C/D layout: N = lane & 15)
  const int ci = mnc / 3;
  const int cd = mnc - 3 * ci;
  const int nodec = (ci == 0) ? n0 : (ci == 1) ? n1 : (ci == 2) ? n2 : n3;
  const int dc = 3 * nodec + cd;

#pragma unroll
  for (int v = 0; v < 8; ++v) {
    const int m = v + (hi ? 8 : 0);  // C/D layout: M = vgpr + 8*lane_half
    if (valid && m < 12) {
      const int mi = m / 3;
      const int md = m - 3 * mi;
      const int nodem = (mi == 0) ? n0 : (mi == 1) ? n1 : (mi == 2) ? n2 : n3;
      const int dr = 3 * nodem + md;
      const float val = vol * C[v] - ((m == mn) ? mterm : 0.f);
      atomicAdd(&As[(size_t)dr * NDOFc + dc], val);
    }
  }
}

// ---------------------------------------------------------------------------
extern "C" void kernel_launch(void* const* d_in, const int* in_sizes, int n_in,
                              void* d_out, int out_size, void* d_ws,
                              size_t ws_size, hipStream_t stream) {
  const float* nodes_f = (const float*)d_in[0];
  const float* nodes_s = (const float*)d_in[1];
  const float* Ep = (const float*)d_in[2];
  const float* nup = (const float*)d_in[3];
  const float* rhop = (const float*)d_in[4];
  const int* fel = (const int*)d_in[5];
  const int* selem = (const int*)d_in[6];

  float* out = (float*)d_out;
  float* Af = out;                               // A_f : 9000 x 9000
  float* As = out + (size_t)N_Fc * N_Fc;         // A_s : 9000 x 9000

  const double PI = 3.14159265358979323846;
  const double OMEGA = 2.0 * PI * 1000.0;
  const float kf2 = (float)((OMEGA / 343.0) * (OMEGA / 343.0));  // (w/c)^2
  const float om2 = (float)(OMEGA * OMEGA);

  const int n4 = (2 * N_Fc * N_Fc) / 4;  // 40.5M float4
  zero_out_kernel<<<8192, 256, 0, stream>>>((f4*)out, n4);
  fluid_assemble_kernel<<<(EFc + 255) / 256, 256, 0, stream>>>(nodes_f, fel, Af,
                                                               kf2);
  // ES = 80000 elements, 8 waves/block, exact grid -> EXEC all-ones for WMMA
  solid_assemble_kernel<<<ESc / 8, 256, 0, stream>>>(nodes_s, selem, Ep, nup,
                                                     rhop, As, om2);
}